// GlobalPIScoreNet_51110110822715
// MI455X (gfx1250) — compile-verified
//
#include <hip/hip_runtime.h>

// ---------------------------------------------------------------------------
// GlobalPIScoreNet forward for MI455X (gfx1250, wave32, WMMA).
// Dense GEMMs -> V_WMMA_F32_16X16X4_F32 (fp32 exactness, matches reference).
// Edge message passing -> per-edge VALU with LDS-staged weights + f32 atomics.
// r2a @ h_res is the HBM-bound piece (480 MB stream ~= 21 us at 23.3 TB/s),
// so its inner loop must be branch-free: clamped addressing instead of
// predicated loads, K unrolled 4x (4 WMMAs per iteration), 16x32 wave tiles.
// ---------------------------------------------------------------------------

typedef __attribute__((ext_vector_type(2))) float v2f;
typedef __attribute__((ext_vector_type(8))) float v8f;

#define N_RES   2000
#define N_ATM   60000
#define E_RES   64000
#define E_ATM   1440000
#define NGRAPH  16
#define NC      32

__device__ __forceinline__ float silu_f(float x) { return x / (1.f + __expf(-x)); }

template<int ACT> __device__ __forceinline__ float act_apply(float x) {
  if constexpr (ACT == 1) {           // elu
    return x > 0.f ? x : (__expf(x) - 1.f);
  } else if constexpr (ACT == 2) {    // silu
    return x / (1.f + __expf(-x));
  } else if constexpr (ACT == 3) {    // gelu (tanh approx, jax default)
    return 0.5f * x * (1.f + tanhf(0.7978845608028654f * (x + 0.044715f * x * x * x)));
  } else {
    return x;
  }
}

__device__ __forceinline__ v8f wmma4(v2f a, v2f b, v8f c) {
  return __builtin_amdgcn_wmma_f32_16x16x4_f32(false, a, false, b, (short)0, c, false, false);
}

// ---------------------------------------------------------------------------
// Generic fp32 WMMA GEMM:  C[M,N] = act(A[M,K] @ W[K,N] + bias) (+ R)
// One wave -> one 16x16 (TWO=false) or 16x32 (TWO=true) output tile.
// Out-of-range rows/cols are CLAMPED for loads (their results are never
// stored), so the K main loop has zero predication. Only the K%4 tail uses
// branch-free cndmask zeroing.
// ---------------------------------------------------------------------------
template<int ACT, bool TWO>
__global__ void gemm_wmma_f32(const float* __restrict__ A, int lda,
                              const float* __restrict__ W, int ldw,
                              const float* __restrict__ bias,
                              const float* __restrict__ R, int ldr,
                              float* __restrict__ C, int ldc,
                              int M, int K, int N) {
  const int lane = threadIdx.x & 31;
  const int wave = threadIdx.x >> 5;
  const int colspan = TWO ? 32 : 16;
  const int tiles_n = (N + colspan - 1) / colspan;
  const int tiles_m = (M + 15) >> 4;
  const int t = blockIdx.x * (blockDim.x >> 5) + wave;
  if (t >= tiles_m * tiles_n) return;           // wave-uniform: EXEC stays full
  const int tm = t / tiles_n;
  const int tn = t - tm * tiles_n;
  const int half = lane >> 4;                   // 0: K+0/1,  1: K+2/3
  const int l16  = lane & 15;
  const int arow  = tm * 16 + l16;
  const int arow_c = arow < M ? arow : (M - 1); // clamp: never stored anyway
  const int bcol0 = tn * colspan + l16;
  const int bcol1 = bcol0 + 16;
  const int bc0 = bcol0 < N ? bcol0 : (N - 1);
  const int bc1 = bcol1 < N ? bcol1 : (N - 1);
  const long abase = (long)arow_c * lda;

  v8f acc0 = {0.f, 0.f, 0.f, 0.f, 0.f, 0.f, 0.f, 0.f};
  v8f acc1 = {0.f, 0.f, 0.f, 0.f, 0.f, 0.f, 0.f, 0.f};

  const int K4 = K & ~3;
  int k = 0;
  // main loop: 4 WMMAs per iteration, no predication at all
  for (; k + 16 <= K4; k += 16) {
    if (k + 80 < K) __builtin_prefetch(&A[abase + k + 80], 0, 0);  // uniform guard
#pragma unroll
    for (int u = 0; u < 4; ++u) {
      const int ka = k + u * 4 + half * 2;
      v2f a, b0;
      a.x  = A[abase + ka];
      a.y  = A[abase + ka + 1];
      b0.x = W[(long)ka * ldw + bc0];
      b0.y = W[(long)(ka + 1) * ldw + bc0];
      acc0 = wmma4(a, b0, acc0);
      if constexpr (TWO) {
        v2f b1;
        b1.x = W[(long)ka * ldw + bc1];
        b1.y = W[(long)(ka + 1) * ldw + bc1];
        acc1 = wmma4(a, b1, acc1);
      }
    }
  }
  for (; k < K4; k += 4) {
    const int ka = k + half * 2;
    v2f a, b0;
    a.x  = A[abase + ka];
    a.y  = A[abase + ka + 1];
    b0.x = W[(long)ka * ldw + bc0];
    b0.y = W[(long)(ka + 1) * ldw + bc0];
    acc0 = wmma4(a, b0, acc0);
    if constexpr (TWO) {
      v2f b1;
      b1.x = W[(long)ka * ldw + bc1];
      b1.y = W[(long)(ka + 1) * ldw + bc1];
      acc1 = wmma4(a, b1, acc1);
    }
  }
  if (k < K) {                                  // K%4 tail: branch-free cndmask
    const int ka  = k + half * 2;
    const int kc0 = ka < K ? ka : (K - 1);
    const int kc1 = ka + 1 < K ? ka + 1 : (K - 1);
    const float ax = A[abase + kc0], ay = A[abase + kc1];
    const float bx0 = W[(long)kc0 * ldw + bc0], by0 = W[(long)kc1 * ldw + bc0];
    v2f a, b0;
    a.x = ka < K ? ax : 0.f;       a.y = ka + 1 < K ? ay : 0.f;
    b0.x = ka < K ? bx0 : 0.f;     b0.y = ka + 1 < K ? by0 : 0.f;
    acc0 = wmma4(a, b0, acc0);
    if constexpr (TWO) {
      const float bx1 = W[(long)kc0 * ldw + bc1], by1 = W[(long)kc1 * ldw + bc1];
      v2f b1;
      b1.x = ka < K ? bx1 : 0.f;   b1.y = ka + 1 < K ? by1 : 0.f;
      acc1 = wmma4(a, b1, acc1);
    }
  }

#pragma unroll
  for (int i = 0; i < 8; ++i) {
    const int row = tm * 16 + i + half * 8;
    if (row < M) {
      if (bcol0 < N) {
        float v = acc0[i];
        if (bias) v += bias[bcol0];
        v = act_apply<ACT>(v);
        if (R) v += R[(long)row * ldr + bcol0];
        C[(long)row * ldc + bcol0] = v;
      }
      if constexpr (TWO) {
        if (bcol1 < N) {
          float v = acc1[i];
          if (bias) v += bias[bcol1];
          v = act_apply<ACT>(v);
          if (R) v += R[(long)row * ldr + bcol1];
          C[(long)row * ldc + bcol1] = v;
        }
      }
    }
  }
}

// ---------------------------------------------------------------------------
// EGNN edge kernel (nf=32, edge_nf=2): per-edge thread.
//   m   = silu(silu([h_s,h_d,r,e] @ we1 + be1) @ we2 + be2)
//   cw  = silu(m @ wc1 + bc1) @ wc2
//   cat[src, 32:64] += m ; xagg[src,0:3] += diff*cw ; xagg[src,3] += 1
// Weights staged in LDS (broadcast reads -> no bank conflicts).
// ---------------------------------------------------------------------------
#define SW_WE1 0
#define SW_BE1 2144
#define SW_WE2 2176
#define SW_BE2 3200
#define SW_WC1 3232
#define SW_BC1 4256
#define SW_WC2 4288
#define SW_TOT 4320

__global__ void egnn_edge(const float* __restrict__ h, const float* __restrict__ x,
                          const int* __restrict__ src, const int* __restrict__ dst,
                          const float* __restrict__ eattr,
                          const float* __restrict__ we1, const float* __restrict__ be1,
                          const float* __restrict__ we2, const float* __restrict__ be2,
                          const float* __restrict__ wc1, const float* __restrict__ bc1,
                          const float* __restrict__ wc2,
                          float* __restrict__ cat, float* __restrict__ xagg, int E) {
  __shared__ float sm[SW_TOT];
  const int bs = blockDim.x;
  for (int i = threadIdx.x; i < 2144; i += bs) sm[SW_WE1 + i] = we1[i];
  for (int i = threadIdx.x; i < 1024; i += bs) { sm[SW_WE2 + i] = we2[i]; sm[SW_WC1 + i] = wc1[i]; }
  for (int i = threadIdx.x; i < 32;   i += bs) {
    sm[SW_BE1 + i] = be1[i]; sm[SW_BE2 + i] = be2[i];
    sm[SW_BC1 + i] = bc1[i]; sm[SW_WC2 + i] = wc2[i];
  }
  __syncthreads();

  const int e = blockIdx.x * bs + threadIdx.x;
  if (e >= E) return;
  const int s = src[e], d = dst[e];
  const float dx = x[s * 3 + 0] - x[d * 3 + 0];
  const float dy = x[s * 3 + 1] - x[d * 3 + 1];
  const float dz = x[s * 3 + 2] - x[d * 3 + 2];
  const float radial = dx * dx + dy * dy + dz * dz;
  const float e0 = eattr[e * 2 + 0], e1 = eattr[e * 2 + 1];

  float acc[32];
#pragma unroll
  for (int c = 0; c < 32; ++c)
    acc[c] = sm[SW_BE1 + c] + radial * sm[SW_WE1 + 64 * 32 + c]
           + e0 * sm[SW_WE1 + 65 * 32 + c] + e1 * sm[SW_WE1 + 66 * 32 + c];

  const float* hs = h + (long)s * NC;
  const float* hd = h + (long)d * NC;
  for (int k = 0; k < 32; ++k) {        // rolled; h reads hit L0/L2
    const float a = hs[k], b = hd[k];
#pragma unroll
    for (int c = 0; c < 32; ++c)
      acc[c] += a * sm[SW_WE1 + k * 32 + c] + b * sm[SW_WE1 + (32 + k) * 32 + c];
  }
  float m1[32];
#pragma unroll
  for (int c = 0; c < 32; ++c) m1[c] = silu_f(acc[c]);

  float m2[32];
#pragma unroll
  for (int c = 0; c < 32; ++c) {
    float sacc = sm[SW_BE2 + c];
#pragma unroll
    for (int k = 0; k < 32; ++k) sacc += m1[k] * sm[SW_WE2 + k * 32 + c];
    m2[c] = silu_f(sacc);
  }
  float cw = 0.f;
#pragma unroll
  for (int c = 0; c < 32; ++c) {
    float sacc = sm[SW_BC1 + c];
#pragma unroll
    for (int k = 0; k < 32; ++k) sacc += m2[k] * sm[SW_WC1 + k * 32 + c];
    cw += silu_f(sacc) * sm[SW_WC2 + c];
  }

  float* crow = cat + (long)s * 64 + 32;
#pragma unroll
  for (int c = 0; c < 32; ++c) atomicAdd(&crow[c], m2[c]);
  atomicAdd(&xagg[s * 4 + 0], dx * cw);
  atomicAdd(&xagg[s * 4 + 1], dy * cw);
  atomicAdd(&xagg[s * 4 + 2], dz * cw);
  atomicAdd(&xagg[s * 4 + 3], 1.0f);
}

// x += xagg/max(cnt,1);  cat[:,0:32] = h   (prepares node-MLP GEMM input)
__global__ void egnn_node_prep(const float* __restrict__ h, float* __restrict__ x,
                               float* __restrict__ cat, const float* __restrict__ xagg,
                               int N) {
  const int n = blockIdx.x * blockDim.x + threadIdx.x;
  if (n >= N) return;
  float c = xagg[n * 4 + 3];
  c = c < 1.f ? 1.f : c;
#pragma unroll
  for (int i = 0; i < 3; ++i) x[n * 3 + i] += xagg[n * 4 + i] / c;
#pragma unroll
  for (int i = 0; i < 32; ++i) cat[(long)n * 64 + i] = h[(long)n * 32 + i];
}

__global__ void fill_zero(float* __restrict__ p, long n) {
  const long i = (long)blockIdx.x * blockDim.x + threadIdx.x;
  if (i < n) p[i] = 0.f;
}

// proj = [ f[:,0:1] , h00(15) , h01(15) , f[:,87:151] ]  -> [N,95]
__global__ void build_proj(const float* __restrict__ f, const float* __restrict__ h00,
                           const float* __restrict__ h01, float* __restrict__ proj, int N) {
  const int n = blockIdx.x * blockDim.x + threadIdx.x;
  if (n >= N) return;
  float* p = proj + (long)n * 95;
  const float* fr = f + (long)n * 151;
  p[0] = fr[0];
#pragma unroll
  for (int i = 0; i < 15; ++i) { p[1 + i] = h00[(long)n * 15 + i]; p[16 + i] = h01[(long)n * 15 + i]; }
#pragma unroll
  for (int i = 0; i < 64; ++i) p[31 + i] = fr[87 + i];
}

__global__ void pool_acc(const float* __restrict__ h, const int* __restrict__ gid,
                         float* __restrict__ sums, float* __restrict__ cnt, int N) {
  const int n = blockIdx.x * blockDim.x + threadIdx.x;
  if (n >= N) return;
  const int g = gid[n];
#pragma unroll
  for (int c = 0; c < 32; ++c) atomicAdd(&sums[g * 32 + c], h[(long)n * 32 + c]);
  atomicAdd(&cnt[g], 1.f);
}

__global__ void pool_div(float* __restrict__ sums, const float* __restrict__ cnt) {
  const int i = blockIdx.x * blockDim.x + threadIdx.x;
  if (i >= NGRAPH * 32) return;
  float c = cnt[i >> 5];
  sums[i] /= (c < 1.f ? 1.f : c);
}

// ---------------------------------------------------------------------------
extern "C" void kernel_launch(void* const* d_in, const int* in_sizes, int n_in,
                              void* d_out, int out_size, void* d_ws, size_t ws_size,
                              hipStream_t stream) {
  (void)in_sizes; (void)n_in; (void)out_size; (void)ws_size;
  auto F = [&](int i) { return (const float*)d_in[i]; };
  auto I = [&](int i) { return (const int*)d_in[i]; };

  // --- param leaves, depth-first insertion order of setup_inputs() pytree ---
  // res: 0 l1_w 1 l1_b 2 l2_w 3 l2_b
  // res.egnn: 4 emb_in_w 5 emb_in_b 6 emb_out_w 7 emb_out_b ; layers at 8, 19
  // proj: 30..39 ; atm_egnn: 40..43 emb, layers at 44, 55 ; cls: 66..71
  // data: 72 res_node_feat 73 res_coords 74 res_src 75 res_dst 76 res_efeat
  //       77 atm_node_feat 78 atm_coords 79 atm_src 80 atm_dst 81 atm_efeat
  //       82 r2a 83 graph_id
  const int RL0 = 8, RL1 = 19, AL0 = 44, AL1 = 55;

  // --- workspace carve (floats), ~68 MB total ---
  float* w = (float*)d_ws;
  size_t off = 0;
  auto carve = [&](size_t n) -> float* { float* p = w + off; off += (n + 63) & ~(size_t)63; return p; };
  float* t35      = carve((size_t)N_RES * 35);
  float* hA       = carve((size_t)N_RES * 32);
  float* hB       = carve((size_t)N_RES * 32);
  float* tmpres   = carve((size_t)N_RES * 32);
  float* xres     = carve((size_t)N_RES * 3);
  float* cat_res  = carve((size_t)N_RES * 64);
  float* xagg_res = carve((size_t)N_RES * 4);
  float* hfin     = carve((size_t)N_RES * 32);
  float* hin      = carve((size_t)N_ATM * 64);   // later aliased as cat_atm
  float* P1       = carve((size_t)N_ATM * 15);
  float* P2       = carve((size_t)N_ATM * 15);
  float* P3       = carve((size_t)N_ATM * 15);
  float* proj95   = carve((size_t)N_ATM * 95);   // later aliased as tmp_atm
  float* hatmA    = carve((size_t)N_ATM * 32);
  float* hatmB    = carve((size_t)N_ATM * 32);
  float* xagg_atm = carve((size_t)N_ATM * 4);
  float* xatm     = carve((size_t)N_ATM * 3);
  float* pooled   = carve(NGRAPH * 32);
  float* pcnt     = carve(NGRAPH);
  float* z1       = carve(NGRAPH * 32);
  float* z2       = carve(NGRAPH * 16);
  float* cat_atm  = hin;
  float* tmp_atm  = proj95;

  auto gemm = [&](const float* A, int lda, const float* W_, int ldw, const float* b,
                  const float* R, int ldr, float* C, int ldc,
                  int M, int K, int N, int act) {
    const bool two = N > 16;
    const int colspan = two ? 32 : 16;
    const int tiles = ((M + 15) / 16) * ((N + colspan - 1) / colspan);
    const int blocks = (tiles + 3) / 4;          // 4 waves / block
    switch (act * 2 + (two ? 1 : 0)) {
      case 0: gemm_wmma_f32<0,false><<<blocks,128,0,stream>>>(A,lda,W_,ldw,b,R,ldr,C,ldc,M,K,N); break;
      case 1: gemm_wmma_f32<0,true ><<<blocks,128,0,stream>>>(A,lda,W_,ldw,b,R,ldr,C,ldc,M,K,N); break;
      case 2: gemm_wmma_f32<1,false><<<blocks,128,0,stream>>>(A,lda,W_,ldw,b,R,ldr,C,ldc,M,K,N); break;
      case 3: gemm_wmma_f32<1,true ><<<blocks,128,0,stream>>>(A,lda,W_,ldw,b,R,ldr,C,ldc,M,K,N); break;
      case 4: gemm_wmma_f32<2,false><<<blocks,128,0,stream>>>(A,lda,W_,ldw,b,R,ldr,C,ldc,M,K,N); break;
      case 5: gemm_wmma_f32<2,true ><<<blocks,128,0,stream>>>(A,lda,W_,ldw,b,R,ldr,C,ldc,M,K,N); break;
      case 6: gemm_wmma_f32<3,false><<<blocks,128,0,stream>>>(A,lda,W_,ldw,b,R,ldr,C,ldc,M,K,N); break;
      default:gemm_wmma_f32<3,true ><<<blocks,128,0,stream>>>(A,lda,W_,ldw,b,R,ldr,C,ldc,M,K,N); break;
    }
  };
  auto zero = [&](float* p, long n) { fill_zero<<<(int)((n + 255) / 256), 256, 0, stream>>>(p, n); };

  auto gcl = [&](int base, const float* hi, float* xcur, float* cat, float* xagg,
                 float* tmp, float* ho, const int* src, const int* dst,
                 const float* efeat, int NN, int EE) {
    zero(cat, (long)NN * 64);
    zero(xagg, (long)NN * 4);
    egnn_edge<<<(EE + 255) / 256, 256, 0, stream>>>(hi, xcur, src, dst, efeat,
        F(base + 0), F(base + 1), F(base + 2), F(base + 3),
        F(base + 8), F(base + 9), F(base + 10), cat, xagg, EE);
    egnn_node_prep<<<(NN + 255) / 256, 256, 0, stream>>>(hi, xcur, cat, xagg, NN);
    gemm(cat, 64, F(base + 4), 32, F(base + 5), nullptr, 0, tmp, 32, NN, 64, 32, 2);  // silu
    gemm(tmp, 32, F(base + 6), 32, F(base + 7), hi, 32, ho, 32, NN, 32, 32, 0);       // + residual
  };

  // ================= Residue graph =================
  gemm(F(72), 35, F(0), 35, F(1), nullptr, 0, t35, 35, N_RES, 35, 35, 1);   // elu
  gemm(t35, 35, F(2), 32, F(3), nullptr, 0, hA, 32, N_RES, 35, 32, 0);
  gemm(hA, 32, F(4), 32, F(5), nullptr, 0, hB, 32, N_RES, 32, 32, 0);       // emb_in
  hipMemcpyAsync(xres, F(73), (size_t)N_RES * 3 * sizeof(float), hipMemcpyDeviceToDevice, stream);
  gcl(RL0, hB, xres, cat_res, xagg_res, tmpres, hA, I(74), I(75), F(76), N_RES, E_RES);
  gcl(RL1, hA, xres, cat_res, xagg_res, tmpres, hB, I(74), I(75), F(76), N_RES, E_RES);
  gemm(hB, 32, F(6), 32, F(7), nullptr, 0, hfin, 32, N_RES, 32, 32, 0);     // emb_out

  // ================= r2a projection (HBM-bound, WMMA) =================
  gemm(F(82), N_RES, hfin, 32, nullptr, nullptr, 0, hin, 64, N_ATM, N_RES, 32, 0);

  // ================= Projection module =================
  const float* f = F(77);
  gemm(f + 1, 151, F(30), 15, F(31), nullptr, 0, P1, 15, N_ATM, 21, 15, 1); // elu
  gemm(P1, 15, F(32), 15, F(33), nullptr, 0, P2, 15, N_ATM, 15, 15, 0);     // h00
  gemm(f + 22, 151, F(34), 15, F(35), nullptr, 0, P1, 15, N_ATM, 65, 15, 1);
  gemm(P1, 15, F(36), 15, F(37), nullptr, 0, P3, 15, N_ATM, 15, 15, 0);     // h01
  build_proj<<<(N_ATM + 255) / 256, 256, 0, stream>>>(f, P2, P3, proj95, N_ATM);
  gemm(proj95, 95, F(38), 32, F(39), nullptr, 0, hin + 32, 64, N_ATM, 95, 32, 1); // h_atm0

  // ================= Atom graph =================
  gemm(hin, 64, F(40), 32, F(41), nullptr, 0, hatmA, 32, N_ATM, 64, 32, 0); // emb_in (hin free)
  hipMemcpyAsync(xatm, F(78), (size_t)N_ATM * 3 * sizeof(float), hipMemcpyDeviceToDevice, stream);
  gcl(AL0, hatmA, xatm, cat_atm, xagg_atm, tmp_atm, hatmB, I(79), I(80), F(81), N_ATM, E_ATM);
  gcl(AL1, hatmB, xatm, cat_atm, xagg_atm, tmp_atm, hatmA, I(79), I(80), F(81), N_ATM, E_ATM);
  gemm(hatmA, 32, F(42), 32, F(43), nullptr, 0, hatmB, 32, N_ATM, 32, 32, 0); // emb_out

  // ================= Pool + classifier =================
  zero(pooled, NGRAPH * 32);
  zero(pcnt, NGRAPH);
  pool_acc<<<(N_ATM + 255) / 256, 256, 0, stream>>>(hatmB, I(83), pooled, pcnt, N_ATM);
  pool_div<<<2, 256, 0, stream>>>(pooled, pcnt);
  gemm(pooled, 32, F(66), 32, F(67), nullptr, 0, z1, 32, NGRAPH, 32, 32, 3); // gelu
  gemm(z1, 32, F(68), 16, F(69), nullptr, 0, z2, 16, NGRAPH, 32, 16, 3);     // gelu
  gemm(z2, 16, F(70), 9, F(71), nullptr, 0, (float*)d_out, 9, NGRAPH, 16, 9, 0);
}